// GraphEncoder_80582176408035
// MI455X (gfx1250) — compile-verified
//
#include <hip/hip_runtime.h>
#include <hip/hip_bf16.h>

// Sizes from the reference
#define BGRAPHS 32
#define NNODES  2048
#define BNODES  65536   // BGRAPHS * NNODES
#define FDIM    128
#define NDIM    16

typedef __attribute__((ext_vector_type(2))) float v2f;
typedef __attribute__((ext_vector_type(8))) float v8f;

// ---------------------------------------------------------------------------
// Kernel 1: h = relu(x @ W1 + b1) @ W2 + b2   (fp32 WMMA 16x16x4)
// One workgroup = 16 rows of h, 8 waves, each wave owns a 16-column tile.
// ---------------------------------------------------------------------------
__global__ __launch_bounds__(256) void node_mlp_kernel(
    const float* __restrict__ x,
    const float* __restrict__ w1, const float* __restrict__ b1,
    const float* __restrict__ w2, const float* __restrict__ b2,
    float* __restrict__ h)
{
    __shared__ float xs[16 * 18];    // 16x16 x-tile, stride 18 (bank-conflict free)
    __shared__ float hs[16 * 132];   // 16x128 hidden tile, stride 132

    const int tid  = threadIdx.x;
    const int wave = tid >> 5;
    const int lane = tid & 31;
    const int row0 = blockIdx.x * 16;

    // stage x tile: 256 threads cover 16x16 exactly
    {
        int r = tid >> 4, c = tid & 15;
        xs[r * 18 + c] = x[(row0 + r) * NDIM + c];
    }
    __syncthreads();

    const int mrow  = lane & 15;          // A-matrix row / D-matrix column index
    const int khalf = (lane >> 4) * 2;    // K sub-offset per lane half
    const int ncol  = wave * 16 + mrow;   // output column owned by this lane

    // ---- GEMM1: [16x16] @ [16x128] tile, K in 4 steps of 4 ----
    float bias = b1[ncol];
    v8f acc = {bias, bias, bias, bias, bias, bias, bias, bias};
    for (int k0 = 0; k0 < NDIM; k0 += 4) {
        v2f a;
        a.x = xs[mrow * 18 + k0 + khalf];
        a.y = xs[mrow * 18 + k0 + khalf + 1];
        v2f b;
        b.x = w1[(k0 + khalf)     * FDIM + ncol];
        b.y = w1[(k0 + khalf + 1) * FDIM + ncol];
        acc = __builtin_amdgcn_wmma_f32_16x16x4_f32(
            false, a, false, b, (short)0, acc, false, false);
    }
    // ReLU, scatter C/D layout (rows r and r+8 per half-wave) into LDS
    for (int i = 0; i < 8; ++i) {
        float v = acc[i] > 0.0f ? acc[i] : 0.0f;
        int r = i + (lane >> 4) * 8;
        hs[r * 132 + ncol] = v;
    }
    __syncthreads();

    // ---- GEMM2: [16x128] @ [128x128] tile, K in 32 steps of 4 ----
    float bias2 = b2[ncol];
    v8f acc2 = {bias2, bias2, bias2, bias2, bias2, bias2, bias2, bias2};
    for (int k0 = 0; k0 < FDIM; k0 += 4) {
        v2f a = *(const v2f*)&hs[mrow * 132 + k0 + khalf];
        v2f b;
        b.x = w2[(k0 + khalf)     * FDIM + ncol];
        b.y = w2[(k0 + khalf + 1) * FDIM + ncol];
        acc2 = __builtin_amdgcn_wmma_f32_16x16x4_f32(
            false, a, false, b, (short)0, acc2, false, false);
    }
    for (int i = 0; i < 8; ++i) {
        int r = i + (lane >> 4) * 8;
        h[(row0 + r) * FDIM + ncol] = acc2[i];
    }
}

// ---------------------------------------------------------------------------
// Kernel 2: per-node score = tanh(h.w / ||w||); orderable key (0 if masked out)
// One wave (32 lanes x float4) per node.
// ---------------------------------------------------------------------------
__device__ __forceinline__ unsigned order_map(float f) {
    unsigned b = __float_as_uint(f);
    return (b & 0x80000000u) ? ~b : (b | 0x80000000u);
}

__global__ __launch_bounds__(256) void score_kernel(
    const float* __restrict__ h, const float* __restrict__ w,
    const int* __restrict__ mask, int use_mask,
    float* __restrict__ s, unsigned* __restrict__ key)
{
    const int wave = threadIdx.x >> 5;
    const int lane = threadIdx.x & 31;
    const int node = blockIdx.x * 8 + wave;

    float4 wv = *(const float4*)&w[lane * 4];
    float4 hv = *(const float4*)&h[(size_t)node * FDIM + lane * 4];
    float wsq = wv.x * wv.x + wv.y * wv.y + wv.z * wv.z + wv.w * wv.w;
    float d   = hv.x * wv.x + hv.y * wv.y + hv.z * wv.z + hv.w * wv.w;
    for (int off = 16; off > 0; off >>= 1) {
        d   += __shfl_xor(d,   off, 32);
        wsq += __shfl_xor(wsq, off, 32);
    }
    if (lane == 0) {
        float sc = tanhf(d / sqrtf(wsq));
        int m = use_mask ? mask[node] : 1;
        s[node]   = sc;
        key[node] = m ? order_map(sc) : 0u;   // masked -> smallest key
    }
}

// ---------------------------------------------------------------------------
// Kernel 3: per-graph radix top-k select, write mask, scale h by sel?score:0
// One workgroup (256 threads) per graph.
// ---------------------------------------------------------------------------
__global__ __launch_bounds__(256) void select_scale_kernel(
    float* __restrict__ h, const float* __restrict__ s,
    const unsigned* __restrict__ key, int* __restrict__ mask, int k)
{
    __shared__ unsigned hist[256];
    __shared__ unsigned sh_prefix;
    __shared__ int sh_kk;
    __shared__ int tieCnt;
    __shared__ float gfac[NNODES];

    const int tid  = threadIdx.x;
    const int base = blockIdx.x * NNODES;

    if (tid == 0) { sh_prefix = 0u; sh_kk = k; tieCnt = 0; }
    __syncthreads();

    // 4-pass MSB-first radix select of the k-th largest key
    for (int p = 3; p >= 0; --p) {
        hist[tid] = 0u;
        __syncthreads();
        unsigned prefix = sh_prefix;
        unsigned hibits = (p == 3) ? 0u : (0xFFFFFFFFu << ((p + 1) * 8));
        for (int n = tid; n < NNODES; n += 256) {
            unsigned u = key[base + n];
            if ((u & hibits) == prefix)
                atomicAdd(&hist[(u >> (p * 8)) & 255], 1u);
        }
        __syncthreads();
        if (tid == 0) {
            int kk = sh_kk;
            int b  = 255;
            for (; b > 0; --b) {
                int c = (int)hist[b];
                if (kk <= c) break;
                kk -= c;
            }
            sh_prefix = prefix | ((unsigned)b << (p * 8));
            sh_kk = kk;
        }
        __syncthreads();   // protects hist from next-pass zeroing, publishes prefix
    }
    const unsigned T  = sh_prefix;   // k-th largest key value
    const int      kk = sh_kk;       // number of ties at T to accept

    // selection + per-node factor
    for (int n = tid; n < NNODES; n += 256) {
        unsigned u = key[base + n];
        int sel = 0;
        if (u > T) sel = 1;
        else if (u == T) {
            int slot = atomicAdd(&tieCnt, 1);
            if (slot < kk) sel = 1;
        }
        mask[base + n] = sel;
        gfac[n] = sel ? s[base + n] : 0.0f;
    }
    __syncthreads();

    // coalesced scaling of h rows
    const size_t hbase = (size_t)base * FDIM;
    for (int idx = tid; idx < NNODES * FDIM; idx += 256) {
        int n = idx >> 7;   // FDIM == 128
        h[hbase + idx] *= gfac[n];
    }
}

// ---------------------------------------------------------------------------
// Kernel 4: per-graph mean over the 256 surviving nodes (others are exact 0)
// ---------------------------------------------------------------------------
__global__ __launch_bounds__(128) void pool_kernel(
    const float* __restrict__ h, float* __restrict__ out)
{
    const int g = blockIdx.x, c = threadIdx.x;
    const float* hp = h + (size_t)g * NNODES * FDIM + c;
    float sum = 0.0f;
    for (int n = 0; n < NNODES; ++n) sum += hp[(size_t)n * FDIM];
    out[g * FDIM + c] = sum * (1.0f / 256.0f);   // KS[2] survivors, static
}

// ---------------------------------------------------------------------------
extern "C" void kernel_launch(void* const* d_in, const int* in_sizes, int n_in,
                              void* d_out, int out_size, void* d_ws, size_t ws_size,
                              hipStream_t stream) {
    (void)in_sizes; (void)n_in; (void)out_size; (void)ws_size;
    const float* x       = (const float*)d_in[0];
    const float* node_w1 = (const float*)d_in[2];
    const float* node_b1 = (const float*)d_in[3];
    const float* node_w2 = (const float*)d_in[4];
    const float* node_b2 = (const float*)d_in[5];
    const float* pool_w  = (const float*)d_in[22];
    // everything else (edge path, eff_*, upd_*) is dead compute in the reference

    char* ws = (char*)d_ws;
    float*    h    = (float*)ws;                                   // 32 MB
    float*    s    = (float*)(ws + (size_t)BNODES * FDIM * 4);     // 256 KB
    unsigned* key  = (unsigned*)((char*)s + (size_t)BNODES * 4);   // 256 KB
    int*      mask = (int*)((char*)key + (size_t)BNODES * 4);      // 256 KB

    node_mlp_kernel<<<BNODES / 16, 256, 0, stream>>>(x, node_w1, node_b1,
                                                     node_w2, node_b2, h);
    const int ks[3] = {1024, 512, 256};
    for (int i = 0; i < 3; ++i) {
        score_kernel<<<BNODES / 8, 256, 0, stream>>>(h, pool_w + i * FDIM,
                                                     mask, i > 0, s, key);
        select_scale_kernel<<<BGRAPHS, 256, 0, stream>>>(h, s, key, mask, ks[i]);
    }
    pool_kernel<<<BGRAPHS, 128, 0, stream>>>(h, (float*)d_out);
}